// MFbpr_62234076119267
// MI455X (gfx1250) — compile-verified
//
#include <hip/hip_runtime.h>
#include <hip/hip_bf16.h>

typedef __attribute__((ext_vector_type(2))) float v2f;
typedef __attribute__((ext_vector_type(8))) float v8f;

#define P1_BLOCKS   128
#define PART_STRIDE 132   // 64 sI + 64 sJ + 1 sq, padded
#define REG_COEF    0.01f
#define INV_LN2     1.4426950408889634f

// ---------------------------------------------------------------------------
// Kernel 1: per-block deterministic partials of s_I, s_J and sum-of-squares.
// Thread t handles chunk c = t&15 (4 floats of the 64-wide row) for row-group
// rg = t>>4; rows are strided deterministically, accumulation order is fixed.
// ---------------------------------------------------------------------------
__global__ void bpr_reduce_partials(const float* __restrict__ EU,
                                    const float* __restrict__ EI,
                                    const int* __restrict__ u,
                                    const int* __restrict__ i,
                                    const int* __restrict__ j,
                                    float* __restrict__ partials, int B)
{
    const int t  = threadIdx.x;      // 256 threads
    const int c  = t & 15;           // float4 chunk within row
    const int rg = t >> 4;           // row group 0..15
    const int p  = blockIdx.x;

    float aIx = 0.f, aIy = 0.f, aIz = 0.f, aIw = 0.f;
    float aJx = 0.f, aJy = 0.f, aJz = 0.f, aJw = 0.f;
    float sq  = 0.f;

    const int rows_per_pass = gridDim.x * 16;
    for (int row = p * 16 + rg; row < B; row += rows_per_pass) {
        const int uu = u[row], ii = i[row], jj = j[row];
        const float4 Uv = *(const float4*)(EU + (size_t)uu * 64 + c * 4);
        const float4 Iv = *(const float4*)(EI + (size_t)ii * 64 + c * 4);
        const float4 Jv = *(const float4*)(EI + (size_t)jj * 64 + c * 4);
        sq += Uv.x*Uv.x + Uv.y*Uv.y + Uv.z*Uv.z + Uv.w*Uv.w;
        sq += Iv.x*Iv.x + Iv.y*Iv.y + Iv.z*Iv.z + Iv.w*Iv.w;
        sq += Jv.x*Jv.x + Jv.y*Jv.y + Jv.z*Jv.z + Jv.w*Jv.w;
        aIx += Iv.x; aIy += Iv.y; aIz += Iv.z; aIw += Iv.w;
        aJx += Jv.x; aJy += Jv.y; aJz += Jv.z; aJw += Jv.w;
    }

    __shared__ float lI[256][4];
    __shared__ float lJ[256][4];
    __shared__ float lsq[256];
    lI[t][0]=aIx; lI[t][1]=aIy; lI[t][2]=aIz; lI[t][3]=aIw;
    lJ[t][0]=aJx; lJ[t][1]=aJy; lJ[t][2]=aJz; lJ[t][3]=aJw;
    lsq[t]=sq;
    __syncthreads();

    // fixed-shape tree reduce for sq (deterministic)
    for (int s = 128; s > 0; s >>= 1) {
        if (t < s) lsq[t] += lsq[t + s];
        __syncthreads();
    }

    float* pblk = partials + (size_t)p * PART_STRIDE;
    if (t < 16) {                 // sum s_I partial for chunk t, fixed order
        float s0=0.f,s1=0.f,s2=0.f,s3=0.f;
        for (int g = 0; g < 16; ++g) {
            s0 += lI[t + 16*g][0]; s1 += lI[t + 16*g][1];
            s2 += lI[t + 16*g][2]; s3 += lI[t + 16*g][3];
        }
        pblk[t*4+0]=s0; pblk[t*4+1]=s1; pblk[t*4+2]=s2; pblk[t*4+3]=s3;
    } else if (t < 32) {          // sum s_J partial for chunk t-16
        const int cc = t - 16;
        float s0=0.f,s1=0.f,s2=0.f,s3=0.f;
        for (int g = 0; g < 16; ++g) {
            s0 += lJ[cc + 16*g][0]; s1 += lJ[cc + 16*g][1];
            s2 += lJ[cc + 16*g][2]; s3 += lJ[cc + 16*g][3];
        }
        pblk[64 + cc*4+0]=s0; pblk[64 + cc*4+1]=s1;
        pblk[64 + cc*4+2]=s2; pblk[64 + cc*4+3]=s3;
    }
    if (t == 0) pblk[128] = lsq[0];
}

// ---------------------------------------------------------------------------
// Kernel 1b: deterministic sequential reduction of block partials.
// ---------------------------------------------------------------------------
__global__ void bpr_sum_partials(const float* __restrict__ partials,
                                 float* __restrict__ sI,
                                 float* __restrict__ sJ,
                                 float* __restrict__ sq)
{
    const int o = threadIdx.x;
    if (o < 129) {
        float s = 0.f;
        for (int p = 0; p < P1_BLOCKS; ++p)
            s += partials[(size_t)p * PART_STRIDE + o];
        if (o < 64)       sI[o] = s;
        else if (o < 128) sJ[o - 64] = s;
        else              *sq = s;
    }
}

// ---------------------------------------------------------------------------
// Kernel 2: y_ui / y_uj via V_WMMA_F32_16X16X4_F32.
// One wave per 16-row tile. A = 16x4 slab of gathered U rows; B = 4x16 with
// column 0 = s_I segment, column 1 = s_J segment, rest zero.
// B is built branch-free: both s-vectors are loaded unconditionally and
// blended with per-lane weights (FMA), so no exec-mask branching occurs
// anywhere in the WMMA loop (EXEC stays all-ones, loads stay vectorized).
// ---------------------------------------------------------------------------
__global__ void bpr_scores_wmma(const float* __restrict__ EU,
                                const int* __restrict__ u,
                                const float* __restrict__ sI,
                                const float* __restrict__ sJ,
                                float* __restrict__ y_ui,
                                float* __restrict__ y_uj,
                                int tiles)
{
    const int lane = threadIdx.x & 31;
    const int tile = blockIdx.x * (blockDim.x >> 5) + (threadIdx.x >> 5);
    if (tile >= tiles) return;               // wave-uniform exit

    const int nrow = lane & 15;              // A: M index, B/D: N index
    const int half = lane >> 4;              // K sub-split (0: K=0,1  1: K=2,3)
    const int row  = tile * 16 + nrow;

    const int uidx = u[row];
    const float2* __restrict__ urow2 = (const float2*)(EU + (size_t)uidx * 64);
    const float2* __restrict__ sI2   = (const float2*)sI;
    const float2* __restrict__ sJ2   = (const float2*)sJ;

    // lane weights for the B-fragment blend (computed once, branch-free)
    const float w0 = (nrow == 0) ? 1.0f : 0.0f;   // selects s_I column (N=0)
    const float w1 = (nrow == 1) ? 1.0f : 0.0f;   // selects s_J column (N=1)

    v8f acc = {};
#pragma unroll
    for (int kk = 0; kk < 16; ++kk) {
        const float2 av  = urow2[kk * 2 + half];
        const float2 iv  = sI2[kk * 2 + half];
        const float2 jv  = sJ2[kk * 2 + half];
        v2f a; a.x = av.x; a.y = av.y;
        v2f b;
        b.x = w0 * iv.x + w1 * jv.x;
        b.y = w0 * iv.y + w1 * jv.y;
        acc = __builtin_amdgcn_wmma_f32_16x16x4_f32(
            /*neg_a=*/false, a, /*neg_b=*/false, b,
            /*c_mod=*/(short)0, acc, /*reuse_a=*/false, /*reuse_b=*/false);
    }

    // D layout: VGPR r, lanes 0-15 -> (M=r, N=lane); lanes 16-31 -> (M=r+8).
    if (nrow < 2) {
        float* __restrict__ dst = (nrow == 0) ? y_ui : y_uj;
        const int mbase = tile * 16 + half * 8;
#pragma unroll
        for (int r = 0; r < 8; ++r) dst[mbase + r] = acc[r];
    }
}

// ---------------------------------------------------------------------------
// Kernel 3: per-block partial sums of log_sigmoid(y_ui - y_uj), stable form.
// ---------------------------------------------------------------------------
__global__ void bpr_logsig(const float* __restrict__ y_ui,
                           const float* __restrict__ y_uj,
                           float* __restrict__ logpart, int B)
{
    const int idx = blockIdx.x * blockDim.x + threadIdx.x;
    float ls = 0.f;
    if (idx < B) {
        const float x = y_ui[idx] - y_uj[idx];
        ls = fminf(x, 0.f) - log1pf(expf(-fabsf(x)));
    }
    __shared__ float l[256];
    l[threadIdx.x] = ls;
    __syncthreads();
    for (int s = 128; s > 0; s >>= 1) {
        if (threadIdx.x < s) l[threadIdx.x] += l[threadIdx.x + s];
        __syncthreads();
    }
    if (threadIdx.x == 0) logpart[blockIdx.x] = l[0];
}

// ---------------------------------------------------------------------------
// Kernel 4: finalize loss = REG*sumsq - logsum/ln(2).
// ---------------------------------------------------------------------------
__global__ void bpr_final(const float* __restrict__ logpart, int nparts,
                          const float* __restrict__ sq,
                          float* __restrict__ loss_out)
{
    if (threadIdx.x == 0 && blockIdx.x == 0) {
        float s = 0.f;
        for (int p = 0; p < nparts; ++p) s += logpart[p];
        *loss_out = REG_COEF * (*sq) - s * INV_LN2;
    }
}

extern "C" void kernel_launch(void* const* d_in, const int* in_sizes, int n_in,
                              void* d_out, int out_size, void* d_ws, size_t ws_size,
                              hipStream_t stream) {
    const float* EU = (const float*)d_in[0];   // embed_user [1e6, 64] f32
    const float* EI = (const float*)d_in[1];   // embed_item [1e6, 64] f32
    const int*   u  = (const int*)d_in[2];     // [B]
    const int*   i  = (const int*)d_in[3];     // [B]
    const int*   j  = (const int*)d_in[4];     // [B]
    const int B = in_sizes[2];                 // 16384

    float* ws       = (float*)d_ws;
    float* partials = ws;                                   // P1_BLOCKS*132
    float* sI       = ws + (size_t)P1_BLOCKS * PART_STRIDE; // 64 (8B aligned)
    float* sJ       = sI + 64;                              // 64
    float* sq       = sJ + 64;                              // 1
    float* logpart  = sq + 8;                               // up to 64+

    float* y_ui = (float*)d_out;
    float* y_uj = y_ui + B;
    float* loss = y_uj + B;

    bpr_reduce_partials<<<P1_BLOCKS, 256, 0, stream>>>(EU, EI, u, i, j, partials, B);
    bpr_sum_partials<<<1, 256, 0, stream>>>(partials, sI, sJ, sq);

    const int tiles = B / 16;                    // B is a multiple of 16
    const int wpb   = 4;                         // 4 waves (128 threads) / block
    bpr_scores_wmma<<<(tiles + wpb - 1) / wpb, wpb * 32, 0, stream>>>(
        EU, u, sI, sJ, y_ui, y_uj, tiles);

    const int lblocks = (B + 255) / 256;
    bpr_logsig<<<lblocks, 256, 0, stream>>>(y_ui, y_uj, logpart, B);
    bpr_final<<<1, 1, 0, stream>>>(logpart, lblocks, sq, loss);
}